// WKS_34342558499025
// MI455X (gfx1250) — compile-verified
//
#include <hip/hip_runtime.h>

#define BATCH 32
#define LEN   512
#define DIM   256
#define UNITS 64
#define SR    10.0f
#define EPS   1e-7f
#define NSPLIT 8   // L-splits for fused ot/oot stage

typedef __attribute__((ext_vector_type(2))) float v2f;
typedef __attribute__((ext_vector_type(8))) float v8f;

// ---------------------------------------------------------------------------
// Kernel 1: M[which][b] = xcT@wc + xsT@ws + w1   ([D,U] per batch, fp32 WMMA)
//   which=0 (branch 1): M1 = x2T@w2 + x1T@w3 + w1
//   which=1 (branch 2): M2 = x1T@w2 + x2T@w3 + w1
// grid: (2*BATCH, DIM/16, UNITS/16), block: 32 (one wave per 16x16 tile)
// ---------------------------------------------------------------------------
__global__ __launch_bounds__(32)
void k_buildM(const float* __restrict__ x1, const float* __restrict__ x2,
              const float* __restrict__ w1, const float* __restrict__ w2,
              const float* __restrict__ w3, float* __restrict__ Mout) {
    const int lane  = threadIdx.x;
    const int ml    = lane & 15;
    const int hi    = lane >> 4;          // 0 = lanes 0-15, 1 = lanes 16-31
    const int which = blockIdx.x & 1;
    const int b     = blockIdx.x >> 1;
    const int d0    = blockIdx.y << 4;
    const int u0    = blockIdx.z << 4;

    const float* xc = (which ? x1 : x2) + (size_t)b * LEN * DIM;  // pairs with w2
    const float* xs = (which ? x2 : x1) + (size_t)b * LEN * DIM;  // pairs with w3

    v8f acc = {0.f,0.f,0.f,0.f,0.f,0.f,0.f,0.f};
    for (int l0 = 0; l0 < LEN; l0 += 4) {
        const int lk = l0 + (hi << 1);    // this half covers K = lk, lk+1
        v2f ac, as, bc, bs;
        // A tiles: A[m=d][k=l] = x[l][d]  (rows d0+ml, both halves)
        ac.x = xc[(size_t)(lk    ) * DIM + d0 + ml];
        ac.y = xc[(size_t)(lk + 1) * DIM + d0 + ml];
        as.x = xs[(size_t)(lk    ) * DIM + d0 + ml];
        as.y = xs[(size_t)(lk + 1) * DIM + d0 + ml];
        // B tiles: B[k=l][n=u] = w[l][u]
        bc.x = w2[(lk    ) * UNITS + u0 + ml];
        bc.y = w2[(lk + 1) * UNITS + u0 + ml];
        bs.x = w3[(lk    ) * UNITS + u0 + ml];
        bs.y = w3[(lk + 1) * UNITS + u0 + ml];
        acc = __builtin_amdgcn_wmma_f32_16x16x4_f32(false, ac, false, bc,
                                                    (short)0, acc, false, false);
        acc = __builtin_amdgcn_wmma_f32_16x16x4_f32(false, as, false, bs,
                                                    (short)0, acc, false, false);
    }
    float* Mb = Mout + (size_t)(which * BATCH + b) * DIM * UNITS;
    #pragma unroll
    for (int j = 0; j < 8; ++j) {
        const int row = d0 + j + hi * 8;   // C/D layout: VGPR j -> rows j / j+8
        Mb[row * UNITS + u0 + ml] = acc[j] + w1[row * UNITS + u0 + ml];
    }
}

// ---------------------------------------------------------------------------
// Kernel 2: t = x @ M  (16 L-rows per wave, full U=64), then
//           a[b,l] = exp(SR * sum_u tanh(t[l,u]) * we[u])
// grid: (2*BATCH, LEN/16), block: 32
// ---------------------------------------------------------------------------
__global__ __launch_bounds__(32)
void k_energy(const float* __restrict__ x1, const float* __restrict__ x2,
              const float* __restrict__ Mws, const float* __restrict__ we,
              float* __restrict__ a_ws) {
    const int lane  = threadIdx.x;
    const int ml    = lane & 15;
    const int hi    = lane >> 4;
    const int which = blockIdx.x >> 5;
    const int b     = blockIdx.x & 31;
    const int l0    = blockIdx.y << 4;

    const float* x = (which ? x2 : x1) + (size_t)b * LEN * DIM;
    const float* M = Mws + (size_t)(which * BATCH + b) * DIM * UNITS;

    v8f acc0 = {0.f,0.f,0.f,0.f,0.f,0.f,0.f,0.f};
    v8f acc1 = acc0, acc2 = acc0, acc3 = acc0;

    for (int d0 = 0; d0 < DIM; d0 += 4) {
        const int dk = d0 + (hi << 1);    // this half covers K = dk, dk+1
        v2f av;                            // A[m=l][k=d] = x[l][d]
        av.x = x[(size_t)(l0 + ml) * DIM + dk];
        av.y = x[(size_t)(l0 + ml) * DIM + dk + 1];
        v2f b0, b1, b2, b3;                // B[k=d][n=u] = M[d][u]
        b0.x = M[(dk    ) * UNITS +  0 + ml];  b0.y = M[(dk + 1) * UNITS +  0 + ml];
        b1.x = M[(dk    ) * UNITS + 16 + ml];  b1.y = M[(dk + 1) * UNITS + 16 + ml];
        b2.x = M[(dk    ) * UNITS + 32 + ml];  b2.y = M[(dk + 1) * UNITS + 32 + ml];
        b3.x = M[(dk    ) * UNITS + 48 + ml];  b3.y = M[(dk + 1) * UNITS + 48 + ml];
        acc0 = __builtin_amdgcn_wmma_f32_16x16x4_f32(false, av, false, b0, (short)0, acc0, false, false);
        acc1 = __builtin_amdgcn_wmma_f32_16x16x4_f32(false, av, false, b1, (short)0, acc1, false, false);
        acc2 = __builtin_amdgcn_wmma_f32_16x16x4_f32(false, av, false, b2, (short)0, acc2, false, false);
        acc3 = __builtin_amdgcn_wmma_f32_16x16x4_f32(false, av, false, b3, (short)0, acc3, false, false);
    }

    const float we0 = we[ 0 + ml], we1 = we[16 + ml];
    const float we2 = we[32 + ml], we3 = we[48 + ml];
    #pragma unroll
    for (int j = 0; j < 8; ++j) {
        float v = tanhf(acc0[j]) * we0 + tanhf(acc1[j]) * we1
                + tanhf(acc2[j]) * we2 + tanhf(acc3[j]) * we3;
        // reduce across the 16 lanes of each half (halves hold rows j / j+8)
        v += __shfl_xor(v, 1);
        v += __shfl_xor(v, 2);
        v += __shfl_xor(v, 4);
        v += __shfl_xor(v, 8);
        if (ml == 0) {
            const int row = l0 + j + hi * 8;
            a_ws[(which * BATCH + b) * LEN + row] = expf(SR * v);
        }
    }
}

// ---------------------------------------------------------------------------
// Kernel 3: S[which*BATCH+b] = sum_l a[b,l]  (deterministic LDS tree)
// grid: 64, block: 256
// ---------------------------------------------------------------------------
__global__ __launch_bounds__(256)
void k_sum(const float* __restrict__ a_ws, float* __restrict__ S_ws) {
    __shared__ float sm[256];
    const int t = threadIdx.x;
    const float* a = a_ws + (size_t)blockIdx.x * LEN;
    sm[t] = a[t] + a[t + 256];
    __syncthreads();
    for (int off = 128; off > 0; off >>= 1) {
        if (t < off) sm[t] += sm[t + off];
        __syncthreads();
    }
    if (t == 0) S_ws[blockIdx.x] = sm[0];
}

// ---------------------------------------------------------------------------
// Kernel 4 (fused): per (split,b,which) block over 64 L-rows:
//   ww = a/(S+eps)  -> write ww (d==0) and ot = x*ww, accumulate partial oot
// grid: (NSPLIT, BATCH, 2), block: 256 (one thread per d)
// ---------------------------------------------------------------------------
__global__ __launch_bounds__(256)
void k_scale_oot(const float* __restrict__ x1, const float* __restrict__ x2,
                 const float* __restrict__ a_ws, const float* __restrict__ S_ws,
                 float* __restrict__ part_ws, float* __restrict__ out) {
    const int split = blockIdx.x, b = blockIdx.y, which = blockIdx.z;
    const int d = threadIdx.x;
    const int bw = which * BATCH + b;
    const float denom = S_ws[bw] + EPS;

    const float* x = (which ? x2 : x1) + ((size_t)b * LEN) * DIM;
    float* ww_out = out + (size_t)which * (BATCH * LEN) + (size_t)b * LEN;
    float* ot_out = out + (size_t)2 * BATCH * LEN
                        + (size_t)which * ((size_t)BATCH * LEN * DIM)
                        + (size_t)b * LEN * DIM;

    const int lbeg = split * (LEN / NSPLIT);
    const int lend = lbeg + (LEN / NSPLIT);
    float s = 0.f;
    for (int l = lbeg; l < lend; ++l) {
        const float ww = a_ws[bw * LEN + l] / denom;
        const float xv = x[(size_t)l * DIM + d];
        const float o  = xv * ww;
        ot_out[(size_t)l * DIM + d] = o;
        if (d == 0) ww_out[l] = ww;
        s += o;
    }
    part_ws[((size_t)bw * NSPLIT + split) * DIM + d] = s;
}

// ---------------------------------------------------------------------------
// Kernel 5: oot[b,d] = sum over NSPLIT partials (fixed order -> deterministic)
// grid: (BATCH, 2), block: 256
// ---------------------------------------------------------------------------
__global__ __launch_bounds__(256)
void k_oot_final(const float* __restrict__ part_ws, float* __restrict__ out) {
    const int b = blockIdx.x, which = blockIdx.y, d = threadIdx.x;
    const int bw = which * BATCH + b;
    const float* p = part_ws + (size_t)bw * NSPLIT * DIM;
    float s = 0.f;
    #pragma unroll
    for (int k = 0; k < NSPLIT; ++k) s += p[(size_t)k * DIM + d];
    float* oot = out + (size_t)2 * BATCH * LEN + (size_t)2 * BATCH * LEN * DIM
                     + (size_t)which * (BATCH * DIM);
    oot[b * DIM + d] = s;
}

// ---------------------------------------------------------------------------
extern "C" void kernel_launch(void* const* d_in, const int* in_sizes, int n_in,
                              void* d_out, int out_size, void* d_ws, size_t ws_size,
                              hipStream_t stream) {
    const float* x1 = (const float*)d_in[0];
    const float* x2 = (const float*)d_in[1];
    const float* w1 = (const float*)d_in[2];
    const float* w2 = (const float*)d_in[3];
    const float* w3 = (const float*)d_in[4];
    const float* we = (const float*)d_in[5];

    float* ws      = (float*)d_ws;
    float* Mws     = ws;                                     // 2*32*256*64 floats (4 MB)
    float* a_ws    = Mws + (size_t)2 * BATCH * DIM * UNITS;  // 2*32*512 floats
    float* S_ws    = a_ws + (size_t)2 * BATCH * LEN;         // 64 floats
    float* part_ws = S_ws + 64;                              // 2*32*8*256 floats (512 KB)
    float* out     = (float*)d_out;

    k_buildM<<<dim3(2 * BATCH, DIM / 16, UNITS / 16), 32, 0, stream>>>(x1, x2, w1, w2, w3, Mws);
    k_energy<<<dim3(2 * BATCH, LEN / 16), 32, 0, stream>>>(x1, x2, Mws, we, a_ws);
    k_sum<<<dim3(64), 256, 0, stream>>>(a_ws, S_ws);
    k_scale_oot<<<dim3(NSPLIT, BATCH, 2), 256, 0, stream>>>(x1, x2, a_ws, S_ws, part_ws, out);
    k_oot_final<<<dim3(BATCH, 2), 256, 0, stream>>>(part_ws, out);
}